// SGNN_8770323218948
// MI455X (gfx1250) — compile-verified
//
#include <hip/hip_runtime.h>
#include <hip/hip_bf16.h>

// ---------------------------------------------------------------------------
// SGNN forward for MI455X (gfx1250, wave32).
// Core compute = per-edge MLPs -> fused 16-edge-tile kernels using
// V_WMMA_F32_16X16X4_F32 (native f32 matrix path; data is f32 in the ref).
// Edge inputs are assembled in LDS (never materialized in HBM); segment
// means are built with global f32 atomics.
// GEMM inner loops are branch-free: compile-time K bounds, clamped-address
// boundary handling (A pad rows are zeroed), no per-load predication.
// SiLU uses the native v_rcp_f32 path instead of the IEEE divide expansion.
// ---------------------------------------------------------------------------

#define DEVINL __device__ __forceinline__

static constexpr int   N_P     = 8192;
static constexpr int   NOBJ    = 16;
static constexpr int   PBLOCK  = N_P / NOBJ;   // 512 particles per object (contiguous)
static constexpr int   H       = 128;
static constexpr int   SDIM    = 16;
static constexpr int   E_INNER = 65536;
static constexpr int   E_INTER = 131072;
static constexpr float ETA     = 0.001f;

typedef __attribute__((ext_vector_type(2))) float v2f;
typedef __attribute__((ext_vector_type(8))) float v8f;

// x * sigmoid(x); v_exp_f32 + v_rcp_f32 + mul (avoids div_scale/div_fixup chain)
DEVINL float silu(float x) {
  return x * __builtin_amdgcn_rcpf(1.0f + __expf(-x));
}
DEVINL v8f zero8() { v8f z = {0.f,0.f,0.f,0.f,0.f,0.f,0.f,0.f}; return z; }

// ---------------------------------------------------------------------------
// 16x16 f32 tile GEMM:  acc += A(16 x K, LDS, lda) * B(K x ?, global, ldb)
//   K  = padded K loop extent (multiple of 4), A rows [KB,K) are ZERO in LDS
//   KB = valid rows of B
//   COLCLAMP: clamp this lane's B column (caller guards the store)
// Main loop is fully unconditional; at most one boundary iteration uses
// clamped addresses (A is zero there, so the product contributes nothing).
// ---------------------------------------------------------------------------
template<int K, int KB, bool COLCLAMP>
DEVINL v8f wmma_f32_tile(const float* __restrict__ A, int lda,
                         const float* __restrict__ B, int ldb,
                         int colBase, int colMax, v8f acc) {
  const int lane = threadIdx.x & 31;
  const int half = lane >> 4;
  const int mr   = lane & 15;
  int col = colBase + mr;
  if constexpr (COLCLAMP) col = (col < colMax) ? col : (colMax - 1);
  constexpr int KMAIN = KB & ~3;
  const float* pa = A + mr * lda + 2 * half;
  const float* pb = B + (2 * half) * ldb + col;
#pragma unroll 4
  for (int k0 = 0; k0 < KMAIN; k0 += 4) {
    v2f a, b;
    a.x = pa[0];  a.y = pa[1];
    b.x = pb[0];  b.y = pb[ldb];
    acc = __builtin_amdgcn_wmma_f32_16x16x4_f32(false, a, false, b,
                                                (short)0, acc, false, false);
    pa += 4; pb += 4 * ldb;
  }
  if constexpr (KMAIN < K) {   // single padded boundary iteration
    const int ka  = KMAIN + 2 * half;
    const int k0c = (ka     < KB) ? ka     : (KB - 1);
    const int k1c = (ka + 1 < KB) ? ka + 1 : (KB - 1);
    v2f a, b;
    a.x = A[mr * lda + ka]; a.y = A[mr * lda + ka + 1];   // zero in pad rows
    b.x = B[k0c * ldb + col]; b.y = B[k1c * ldb + col];
    acc = __builtin_amdgcn_wmma_f32_16x16x4_f32(false, a, false, b,
                                                (short)0, acc, false, false);
  }
  return acc;
}

// Dual-B variant: shares the A stream between two GEMMs (msg + coef paths).
template<int K, int KB>
DEVINL void wmma_f32_tile_dual(const float* __restrict__ A, int lda,
                               const float* __restrict__ B0,
                               const float* __restrict__ B1, int ldb,
                               int colBase, v8f& acc0, v8f& acc1) {
  const int lane = threadIdx.x & 31;
  const int half = lane >> 4;
  const int mr   = lane & 15;
  const int col  = colBase + mr;
  constexpr int KMAIN = KB & ~3;
  const float* pa  = A  + mr * lda + 2 * half;
  const float* pb0 = B0 + (2 * half) * ldb + col;
  const float* pb1 = B1 + (2 * half) * ldb + col;
#pragma unroll 4
  for (int k0 = 0; k0 < KMAIN; k0 += 4) {
    v2f a, b0, b1;
    a.x  = pa[0];   a.y  = pa[1];
    b0.x = pb0[0];  b0.y = pb0[ldb];
    b1.x = pb1[0];  b1.y = pb1[ldb];
    acc0 = __builtin_amdgcn_wmma_f32_16x16x4_f32(false, a, false, b0,
                                                 (short)0, acc0, false, false);
    acc1 = __builtin_amdgcn_wmma_f32_16x16x4_f32(false, a, false, b1,
                                                 (short)0, acc1, false, false);
    pa += 4; pb0 += 4 * ldb; pb1 += 4 * ldb;
  }
  if constexpr (KMAIN < K) {
    const int ka  = KMAIN + 2 * half;
    const int k0c = (ka     < KB) ? ka     : (KB - 1);
    const int k1c = (ka + 1 < KB) ? ka + 1 : (KB - 1);
    v2f a, b0, b1;
    a.x  = A[mr * lda + ka];    a.y  = A[mr * lda + ka + 1];
    b0.x = B0[k0c * ldb + col]; b0.y = B0[k1c * ldb + col];
    b1.x = B1[k0c * ldb + col]; b1.y = B1[k1c * ldb + col];
    acc0 = __builtin_amdgcn_wmma_f32_16x16x4_f32(false, a, false, b0,
                                                 (short)0, acc0, false, false);
    acc1 = __builtin_amdgcn_wmma_f32_16x16x4_f32(false, a, false, b1,
                                                 (short)0, acc1, false, false);
  }
}

// ---------------------------------------------------------------------------
// Fused edge-layer / edge-readout kernel.
//   FN: node vector channels, FE: edge vector channels, SE: edge scalar dim
//   XDIFF: edge vectors are x[src]-x[dst];  READOUT: pool by obj-pair, out=KOUT
// Per edge: F=[f_src|f_dst|e_f] (3 x FT), inp = [Gram(F) | s_src | s_dst | e_s],
// m_s = MLP(inp), coef = MLP(inp), m_f = F @ coef; accumulates w-weighted sums
// into acc_s/acc_f/acc_cnt keyed by pool index.
// ---------------------------------------------------------------------------
template<int FN, int FE, int SE, bool XDIFF, bool READOUT, int KOUT>
__global__ __launch_bounds__(256) void
edge_layer_kernel(const float* __restrict__ xp,
                  const float* __restrict__ edge_f,   // [E,3,FE] if !XDIFF
                  const float* __restrict__ edge_s,   // [E,SE] if SE>0
                  const float* __restrict__ fnode,    // [n,3,FN]
                  const float* __restrict__ snode,    // [n,H]
                  const int*   __restrict__ esrc,
                  const int*   __restrict__ edst,
                  const float* __restrict__ w,        // [E] or nullptr (ones)
                  const int*   __restrict__ obj_id,   // used if READOUT
                  const float* __restrict__ Wm1, const float* __restrict__ bm1,
                  const float* __restrict__ Wm2, const float* __restrict__ bm2,
                  const float* __restrict__ Wc1, const float* __restrict__ bc1,
                  const float* __restrict__ Wc2, const float* __restrict__ bc2,
                  float* __restrict__ acc_s,    // [npool,H]
                  float* __restrict__ acc_f,    // [npool,3*CO]
                  float* __restrict__ acc_cnt)  // [npool]
{
  constexpr int FT   = 2 * FN + FE;
  constexpr int CO   = READOUT ? KOUT : FN;
  constexpr int DIN  = FT * FT + 2 * H + SE;
  constexpr int DINP = (DIN + 3) & ~3;
  constexpr int OUT2 = FT * CO;

  __shared__ float s_inp[16][DINP];
  __shared__ float s_F[16][3][FT];
  __shared__ float s_hidm[16][H];
  __shared__ float s_hidc[16][H];
  __shared__ float s_coef[16][OUT2];
  __shared__ int   s_src[16], s_dst[16], s_pool[16];
  __shared__ float s_w[16];

  const int tid  = threadIdx.x;
  const int e0   = blockIdx.x * 16;
  const int wave = tid >> 5;
  const int lane = tid & 31;
  const int nn   = lane & 15;
  const int mh   = lane >> 4;

  if (tid < 16) {
    const int e  = e0 + tid;
    const int ss = esrc[e];
    const int dd = edst[e];
    s_src[tid] = ss;
    s_dst[tid] = dd;
    if constexpr (READOUT) s_pool[tid] = obj_id[ss] * NOBJ + obj_id[dd];
    else                   s_pool[tid] = ss;
    s_w[tid] = w ? w[e] : 1.0f;
  }
  __syncthreads();

  // ---- build F (16 x 3 x FT) ----
  for (int i = tid; i < 16 * 3 * FT; i += 256) {
    const int e = i / (3 * FT);
    const int r = i - e * (3 * FT);
    const int c = r / FT;
    const int j = r - c * FT;
    const int ss = s_src[e], dd = s_dst[e];
    float v;
    if (j < FN)          v = fnode[(ss * 3 + c) * FN + j];
    else if (j < 2 * FN) v = fnode[(dd * 3 + c) * FN + (j - FN)];
    else {
      if constexpr (XDIFF) v = xp[ss * 3 + c] - xp[dd * 3 + c];
      else                 v = edge_f[((e0 + e) * 3 + c) * FE + (j - 2 * FN)];
    }
    s_F[e][c][j] = v;
  }
  __syncthreads();

  // ---- Gram invariants + scalar features -> s_inp ----
  for (int i = tid; i < 16 * FT * FT; i += 256) {
    const int e = i / (FT * FT);
    const int r = i - e * (FT * FT);
    const int a = r / FT, b = r - a * FT;
    float v = 0.f;
#pragma unroll
    for (int c = 0; c < 3; ++c) v += s_F[e][c][a] * s_F[e][c][b];
    s_inp[e][r] = v;
  }
  for (int i = tid; i < 16 * H; i += 256) {
    const int e = i >> 7, j = i & (H - 1);
    s_inp[e][FT * FT + j]     = snode[s_src[e] * H + j];
    s_inp[e][FT * FT + H + j] = snode[s_dst[e] * H + j];
  }
  if constexpr (SE > 0) {
    for (int i = tid; i < 16 * SE; i += 256) {
      const int e = i / SE, j = i - e * SE;
      s_inp[e][FT * FT + 2 * H + j] = edge_s[(e0 + e) * SE + j];
    }
  }
  if constexpr (DINP > DIN) {
    for (int i = tid; i < 16 * (DINP - DIN); i += 256) {
      const int e = i / (DINP - DIN), j = i - e * (DINP - DIN);
      s_inp[e][DIN + j] = 0.f;
    }
  }
  __syncthreads();

  // ---- fused GEMM1: hidm = silu(inp@Wm1+bm1), hidc = silu(inp@Wc1+bc1) ----
  {
    v8f am = zero8(), ac = zero8();
    wmma_f32_tile_dual<DINP, DIN>(&s_inp[0][0], DINP, Wm1, Wc1, H,
                                  wave * 16, am, ac);
    const int col = wave * 16 + nn;
    const float bbm = bm1[col];
    const float bbc = bc1[col];
#pragma unroll
    for (int r = 0; r < 8; ++r) {
      s_hidm[r + 8 * mh][col] = silu(am[r] + bbm);
      s_hidc[r + 8 * mh][col] = silu(ac[r] + bbc);
    }
  }
  __syncthreads();

  // ---- m_s = hidm @ Wm2 + bm2 -> weighted atomic segment sum ----
  {
    v8f acc = zero8();
    acc = wmma_f32_tile<H, H, false>(&s_hidm[0][0], H, Wm2, H, wave * 16, H, acc);
    const int col = wave * 16 + nn;
    const float bb = bm2[col];
#pragma unroll
    for (int r = 0; r < 8; ++r) {
      const int m = r + 8 * mh;
      atomicAdd(&acc_s[s_pool[m] * H + col], (acc[r] + bb) * s_w[m]);
    }
  }

  // ---- coef = hidc @ Wc2 + bc2  (OUT2 columns) ----
  {
    constexpr int TILES2 = (OUT2 + 15) / 16;
    if (wave < TILES2) {
      v8f acc = zero8();
      acc = wmma_f32_tile<H, H, true>(&s_hidc[0][0], H, Wc2, OUT2,
                                      wave * 16, OUT2, acc);
      const int col = wave * 16 + nn;
      if (col < OUT2) {
        const float bb = bc2[col];
#pragma unroll
        for (int r = 0; r < 8; ++r) s_coef[r + 8 * mh][col] = acc[r] + bb;
      }
    }
  }
  __syncthreads();

  // ---- m_f = F @ coef -> weighted atomic segment sum; count ----
  for (int i = tid; i < 16 * 3 * CO; i += 256) {
    const int e = i / (3 * CO);
    const int r = i - e * (3 * CO);
    const int c = r / CO, k = r - c * CO;
    float v = 0.f;
#pragma unroll
    for (int j = 0; j < FT; ++j) v += s_F[e][c][j] * s_coef[e][j * CO + k];
    atomicAdd(&acc_f[s_pool[e] * (3 * CO) + r], v * s_w[e]);
  }
  if (tid < 16) atomicAdd(&acc_cnt[s_pool[tid]], s_w[tid]);
}

// ---------------------------------------------------------------------------
// Node update: s += MLP_upd([s | mean_s]),  f += mean_f   (16 nodes / block)
// ---------------------------------------------------------------------------
template<int FN>
__global__ __launch_bounds__(256) void
node_upd_kernel(float* __restrict__ fnode, float* __restrict__ snode,
                const float* __restrict__ num_s, const float* __restrict__ num_f,
                const float* __restrict__ cnt,
                const float* __restrict__ Wu1, const float* __restrict__ bu1,
                const float* __restrict__ Wu2, const float* __restrict__ bu2)
{
  __shared__ float s_inp[16][2 * H];
  __shared__ float s_hid[16][H];
  const int tid = threadIdx.x;
  const int n0  = blockIdx.x * 16;
  const int wave = tid >> 5, lane = tid & 31, nn = lane & 15, mh = lane >> 4;

  for (int i = tid; i < 16 * H; i += 256) {
    const int m = i >> 7, j = i & (H - 1);
    const int node = n0 + m;
    const float den = fmaxf(cnt[node], 1.0f);
    s_inp[m][j]     = snode[node * H + j];
    s_inp[m][H + j] = num_s[node * H + j] / den;
  }
  __syncthreads();
  {
    v8f acc = zero8();
    acc = wmma_f32_tile<2 * H, 2 * H, false>(&s_inp[0][0], 2 * H, Wu1, H,
                                             wave * 16, H, acc);
    const int col = wave * 16 + nn;
    const float bb = bu1[col];
#pragma unroll
    for (int r = 0; r < 8; ++r) s_hid[r + 8 * mh][col] = silu(acc[r] + bb);
  }
  __syncthreads();
  {
    v8f acc = zero8();
    acc = wmma_f32_tile<H, H, false>(&s_hid[0][0], H, Wu2, H, wave * 16, H, acc);
    const int col = wave * 16 + nn;
    const float bb = bu2[col];
#pragma unroll
    for (int r = 0; r < 8; ++r) {
      const int m = r + 8 * mh;
      snode[(n0 + m) * H + col] += acc[r] + bb;
    }
  }
  for (int i = tid; i < 16 * 3 * FN; i += 256) {
    const int m = i / (3 * FN), r = i - m * (3 * FN);
    const int node = n0 + m;
    const float den = fmaxf(cnt[node], 1.0f);
    fnode[node * 3 * FN + r] += num_f[node * 3 * FN + r] / den;
  }
}

// ---------------------------------------------------------------------------
// Small helper kernels
// ---------------------------------------------------------------------------
__global__ void k_embed(const float* __restrict__ xp, const float* __restrict__ hp,
                        const float* __restrict__ W, const float* __restrict__ b,
                        float* __restrict__ s_emb) {
  const int g = blockIdx.x * 256 + threadIdx.x;       // N_P * H threads
  const int node = g >> 7, col = g & (H - 1);
  float acc = b[col];
  for (int k = 0; k < SDIM; ++k) {
    float h = hp[node * SDIM + k];
    if (k == SDIM - 1 && xp[node * 3 + 1] < 0.1f) h = 1.0f;
    acc += h * W[k * H + col];
  }
  s_emb[g] = acc;
}

__global__ __launch_bounds__(256) void
k_objpool(const float* __restrict__ xp, const float* __restrict__ vp,
          const float* __restrict__ s_emb, float* __restrict__ f_o,
          float* __restrict__ s_o) {
  __shared__ float red[256];
  const int o = blockIdx.x, tid = threadIdx.x;
  const int p0 = o * PBLOCK;
  for (int ch = 0; ch < 6 + H; ++ch) {
    float v = 0.f;
    for (int p = tid; p < PBLOCK; p += 256) {
      const int pi = p0 + p;
      if (ch < 6) {
        const int c = ch >> 1;
        v += (ch & 1) ? vp[pi * 3 + c] : xp[pi * 3 + c];
      } else {
        v += s_emb[pi * H + (ch - 6)];
      }
    }
    red[tid] = v; __syncthreads();
    for (int s = 128; s > 0; s >>= 1) { if (tid < s) red[tid] += red[tid + s]; __syncthreads(); }
    if (tid == 0) {
      if (ch < 6) f_o[o * 6 + ch] = red[0] / (float)PBLOCK;     // [o][c][x|v]
      else        s_o[o * H + (ch - 6)] = red[0] * ETA;
    }
    __syncthreads();
  }
}

// out = concat(Aobj[obj_id[node]], Bn[node]) @ W + b   (W: [2H,H])
__global__ void k_lin_cat(const float* __restrict__ Aobj, const float* __restrict__ Bn,
                          const int* __restrict__ obj_id,
                          const float* __restrict__ W, const float* __restrict__ b,
                          float* __restrict__ out) {
  const int g = blockIdx.x * 256 + threadIdx.x;       // N_P * H threads
  const int node = g >> 7, col = g & (H - 1);
  const float* ar = Aobj + obj_id[node] * H;
  const float* br = Bn + node * H;
  float acc = b[col];
  for (int k = 0; k < H; ++k) acc += ar[k] * W[k * H + col];
  for (int k = 0; k < H; ++k) acc += br[k] * W[(H + k) * H + col];
  out[g] = acc;
}

__global__ void k_fpinit(const float* __restrict__ xp, const float* __restrict__ vp,
                         const float* __restrict__ f_o, const int* __restrict__ obj_id,
                         float* __restrict__ f_p) {
  const int g = blockIdx.x * 256 + threadIdx.x;       // N_P * 9
  const int node = g / 9, r = g - node * 9;
  const int c = r / 3, j = r - c * 3;
  const int o = obj_id[node];
  float v;
  if (j == 0)      v = xp[node * 3 + c] - f_o[o * 6 + c * 2 + 0];
  else if (j == 1) v = vp[node * 3 + c] - f_o[o * 6 + c * 2 + 1];
  else             v = vp[node * 3 + c];
  f_p[g] = v;
}

__global__ void k_make_eio(int* __restrict__ s, int* __restrict__ d) {
  const int p = threadIdx.x;
  s[p] = p / NOBJ;
  d[p] = p % NOBJ;
}

__global__ void k_objprep(const float* __restrict__ eaf, const float* __restrict__ eas,
                          const float* __restrict__ cnt_o, const float* __restrict__ f_o,
                          float* __restrict__ ea_o_f, float* __restrict__ ea_o_s,
                          float* __restrict__ w_o, float* __restrict__ f_obj) {
  const int g = blockIdx.x * 256 + threadIdx.x;
  const int T1 = 256 * 6, T2 = T1 + 256 * H, T3 = T2 + 256, T4 = T3 + NOBJ * 3;
  if (g < T1) {
    const int p = g / 6;
    ea_o_f[g] = eaf[g] / fmaxf(cnt_o[p], 1.0f);
  } else if (g < T2) {
    const int q = g - T1;
    const int p = q >> 7;
    ea_o_s[q] = eas[q] / fmaxf(cnt_o[p], 1.0f);
  } else if (g < T3) {
    const int p = g - T2;
    w_o[p] = (cnt_o[p] > 0.f) ? 1.f : 0.f;
  } else if (g < T4) {
    const int r = g - T3;
    const int o = r / 3, c = r - o * 3;
    f_obj[r] = f_o[o * 6 + c * 2 + 1];                 // f_o[...,1:] -> velocity chan
  }
}

__global__ void k_fp2init(const float* __restrict__ f_obj, const float* __restrict__ f_p,
                          const int* __restrict__ obj_id, float* __restrict__ f_p2) {
  const int g = blockIdx.x * 256 + threadIdx.x;       // N_P * 12
  const int node = g / 12, r = g - node * 12;
  const int c = r >> 2, j = r & 3;
  f_p2[g] = (j == 0) ? f_obj[obj_id[node] * 3 + c]
                     : f_p[node * 9 + c * 3 + (j - 1)];
}

__global__ void k_final(const float* __restrict__ f_p2, const int* __restrict__ obj_id,
                        const int* __restrict__ obj_type, float* __restrict__ out) {
  const int g = blockIdx.x * 256 + threadIdx.x;       // N_P * 3
  const int node = g / 3, c = g - node * 3;
  const int ch = (obj_type[obj_id[node]] == 0) ? 0 : 1;
  out[g] = f_p2[node * 12 + c * 4 + ch];
}

// ---------------------------------------------------------------------------
// Orchestration
// ---------------------------------------------------------------------------
extern "C" void kernel_launch(void* const* d_in, const int* in_sizes, int n_in,
                              void* d_out, int out_size, void* d_ws, size_t ws_size,
                              hipStream_t stream) {
  if (n_in < 94) return;

  const float* xp       = (const float*)d_in[0];
  const float* vp       = (const float*)d_in[1];
  const float* hp       = (const float*)d_in[2];
  const int*   obj_id   = (const int*)d_in[3];
  const int*   obj_type = (const int*)d_in[4];
  const int*   ei_inner = (const int*)d_in[5];   // [2, E_INNER] : row0=src, row1=dst
  const int*   ei_inter = (const int*)d_in[6];   // [2, E_INTER]
  const float* P[94];
  for (int i = 8; i < 94; ++i) P[i] = (const float*)d_in[i];

  // ---- workspace carve ----
  float* W = (float*)d_ws;
  size_t o = 0;
  auto alloc = [&](size_t n) { float* p = W + o; o += n; return p; };
  float* s_emb  = alloc((size_t)N_P * H);
  float* s_p1   = alloc((size_t)N_P * H);
  float* s_w    = alloc((size_t)N_P * H);
  float* s_p2   = alloc((size_t)N_P * H);
  float* num_s  = alloc((size_t)N_P * H);
  float* f_p    = alloc((size_t)N_P * 9);
  float* f_w    = alloc((size_t)N_P * 9);
  float* f_p2   = alloc((size_t)N_P * 12);
  float* num_f  = alloc((size_t)N_P * 12);
  float* cnt    = alloc((size_t)N_P);
  float* f_o    = alloc(NOBJ * 6);
  float* s_o    = alloc(NOBJ * H);
  float* f_obj  = alloc(NOBJ * 3);
  float* s_obj  = alloc(NOBJ * H);
  float* eaf    = alloc(256 * 6);
  float* eas    = alloc(256 * H);
  float* cnt_o  = alloc(256);
  float* ea_o_f = alloc(256 * 6);
  float* ea_o_s = alloc(256 * H);
  float* w_o    = alloc(256);
  int*   eio_s  = (int*)alloc(256);
  int*   eio_d  = (int*)alloc(256);
  if (ws_size < o * sizeof(float)) return;

  float* out = (float*)d_out;

  // ---- stage 0: embeddings + object pooling + particle frame ----
  k_embed<<<(N_P * H) / 256, 256, 0, stream>>>(xp, hp, P[8], P[9], s_emb);
  k_objpool<<<NOBJ, 256, 0, stream>>>(xp, vp, s_emb, f_o, s_o);
  k_lin_cat<<<(N_P * H) / 256, 256, 0, stream>>>(s_o, s_emb, obj_id, P[10], P[11], s_p1);
  k_fpinit<<<(N_P * 9) / 256, 256, 0, stream>>>(xp, vp, f_o, obj_id, f_p);

  // working copies for the 'local' net (originals are reused by 'o2p')
  hipMemcpyAsync(f_w, f_p, (size_t)N_P * 9 * sizeof(float), hipMemcpyDeviceToDevice, stream);
  hipMemcpyAsync(s_w, s_p1, (size_t)N_P * H * sizeof(float), hipMemcpyDeviceToDevice, stream);

  const int* src_t = ei_inter;
  const int* dst_t = ei_inter + E_INTER;
  const int* src_i = ei_inner;
  const int* dst_i = ei_inner + E_INNER;

  // ---- 'local' mp_net (FN=3, FE=1, din=305), 2 layers ----
  for (int l = 0; l < 2; ++l) {
    const int b = 14 + 12 * l;   // coef @b, msg_s @b+4, upd @b+8
    hipMemsetAsync(num_s, 0, (size_t)N_P * H * sizeof(float), stream);
    hipMemsetAsync(num_f, 0, (size_t)N_P * 9 * sizeof(float), stream);
    hipMemsetAsync(cnt,   0, (size_t)N_P * sizeof(float), stream);
    edge_layer_kernel<3, 1, 0, true, false, 0><<<E_INTER / 16, 256, 0, stream>>>(
        xp, nullptr, nullptr, f_w, s_w, src_t, dst_t, nullptr, nullptr,
        P[b + 4], P[b + 5], P[b + 6], P[b + 7],   // msg_s
        P[b + 0], P[b + 1], P[b + 2], P[b + 3],   // coef
        num_s, num_f, cnt);
    node_upd_kernel<3><<<N_P / 16, 256, 0, stream>>>(
        f_w, s_w, num_s, num_f, cnt, P[b + 8], P[b + 9], P[b + 10], P[b + 11]);
  }
  // ---- 'local' readout -> pooled onto object-pair graph ----
  hipMemsetAsync(eas,   0, 256 * H * sizeof(float), stream);
  hipMemsetAsync(eaf,   0, 256 * 6 * sizeof(float), stream);
  hipMemsetAsync(cnt_o, 0, 256 * sizeof(float), stream);
  edge_layer_kernel<3, 1, 0, true, true, 2><<<E_INTER / 16, 256, 0, stream>>>(
      xp, nullptr, nullptr, f_w, s_w, src_t, dst_t, nullptr, obj_id,
      P[42], P[43], P[44], P[45],   // readout 's'
      P[38], P[39], P[40], P[41],   // readout 'coef'
      eas, eaf, cnt_o);
  k_objprep<<<136, 256, 0, stream>>>(eaf, eas, cnt_o, f_o, ea_o_f, ea_o_s, w_o, f_obj);
  hipMemcpyAsync(s_obj, s_o, NOBJ * H * sizeof(float), hipMemcpyDeviceToDevice, stream);
  k_make_eio<<<1, 256, 0, stream>>>(eio_s, eio_d);

  // ---- 'obj' mp_net (FN=1, FE=2, SE=128, din=400), 2 layers, weighted ----
  for (int l = 0; l < 2; ++l) {
    const int b = 70 + 12 * l;
    hipMemsetAsync(num_s, 0, NOBJ * H * sizeof(float), stream);
    hipMemsetAsync(num_f, 0, NOBJ * 3 * sizeof(float), stream);
    hipMemsetAsync(cnt,   0, NOBJ * sizeof(float), stream);
    edge_layer_kernel<1, 2, H, false, false, 0><<<256 / 16, 256, 0, stream>>>(
        nullptr, ea_o_f, ea_o_s, f_obj, s_obj, eio_s, eio_d, w_o, nullptr,
        P[b + 4], P[b + 5], P[b + 6], P[b + 7],
        P[b + 0], P[b + 1], P[b + 2], P[b + 3],
        num_s, num_f, cnt);
    node_upd_kernel<1><<<1, 256, 0, stream>>>(
        f_obj, s_obj, num_s, num_f, cnt, P[b + 8], P[b + 9], P[b + 10], P[b + 11]);
  }

  // ---- object -> particle refinement ('o2p': FN=4, FE=1, din=337) ----
  k_lin_cat<<<(N_P * H) / 256, 256, 0, stream>>>(s_obj, s_p1, obj_id, P[12], P[13], s_p2);
  k_fp2init<<<(N_P * 12) / 256, 256, 0, stream>>>(f_obj, f_p, obj_id, f_p2);
  for (int l = 0; l < 2; ++l) {
    const int b = 46 + 12 * l;
    hipMemsetAsync(num_s, 0, (size_t)N_P * H * sizeof(float), stream);
    hipMemsetAsync(num_f, 0, (size_t)N_P * 12 * sizeof(float), stream);
    hipMemsetAsync(cnt,   0, (size_t)N_P * sizeof(float), stream);
    edge_layer_kernel<4, 1, 0, true, false, 0><<<E_INNER / 16, 256, 0, stream>>>(
        xp, nullptr, nullptr, f_p2, s_p2, src_i, dst_i, nullptr, nullptr,
        P[b + 4], P[b + 5], P[b + 6], P[b + 7],
        P[b + 0], P[b + 1], P[b + 2], P[b + 3],
        num_s, num_f, cnt);
    node_upd_kernel<4><<<N_P / 16, 256, 0, stream>>>(
        f_p2, s_p2, num_s, num_f, cnt, P[b + 8], P[b + 9], P[b + 10], P[b + 11]);
  }

  // ---- predictor: rigid -> channel 0, else channel 1 ----
  k_final<<<(N_P * 3) / 256, 256, 0, stream>>>(f_p2, obj_id, obj_type, out);
}